// MyNet_17222818857298
// MI455X (gfx1250) — compile-verified
//
#include <hip/hip_runtime.h>
#include <math.h>

// Row-wise mean/variance normalization (LayerNorm w/o affine):
//   y[r, :] = (x[r, :] - mean_r) * rsqrt(var_r + eps)
// Shapes: 32768 rows x 1024 fp32 columns. Pure bandwidth problem:
// 128 MiB in + 128 MiB out -> ~11.5 us floor at 23.3 TB/s.
//
// CDNA5 strategy: one wave32 per 4KB row. Stage the row into LDS with the
// gfx1250 async global->LDS data mover (fire-and-forget, tracked by ASYNCcnt,
// ordered by s_wait_asynccnt -- no workgroup barrier needed since producer
// and consumer are the same wave), then reduce with wave32 shuffles.

#define WAVE            32
#define ROWS_PER_BLOCK  8          // 8 waves per 256-thread block
#define ROW_W           1024       // floats per row
#define CHUNKS          8          // ROW_W / (WAVE * 4 floats per b128)
#define ROW_EPS         1e-5f

typedef __attribute__((ext_vector_type(4))) float f32x4;

// Types matching the async-builtin signature discovered from the round-1
// diagnostic: param is "int __vector(4) * __device__" (AS1 global).
typedef int v4i_t __attribute__((vector_size(16)));
typedef __attribute__((address_space(1))) v4i_t* gbl_v4i_ptr;
typedef __attribute__((address_space(3))) v4i_t* lds_v4i_ptr;

__global__ __launch_bounds__(ROWS_PER_BLOCK * WAVE)
void rownorm_async_kernel(const float* __restrict__ x,
                          float* __restrict__ y,
                          int nrows) {
    __shared__ float smem[ROWS_PER_BLOCK][ROW_W];

    const int tid  = threadIdx.x;
    const int lane = tid & (WAVE - 1);
    const int wv   = tid >> 5;                       // wave id within block
    const int row  = blockIdx.x * ROWS_PER_BLOCK + wv;
    if (row >= nrows) return;                        // uniform per wave

    const float* src = x + (size_t)row * ROW_W;

    // ---- async global -> LDS stage of the whole 4KB row (8 x b128/lane) ----
    #pragma unroll
    for (int c = 0; c < CHUNKS; ++c) {
        const int col = c * (WAVE * 4) + lane * 4;   // 512B contiguous per op
#if __has_builtin(__builtin_amdgcn_global_load_async_to_lds_b128)
        __builtin_amdgcn_global_load_async_to_lds_b128(
            (gbl_v4i_ptr)(src + col),
            (lds_v4i_ptr)&smem[wv][col],
            /*offset=*/0, /*cpol=*/0);
#else
        unsigned lds_addr = (unsigned)(unsigned long long)
            (__attribute__((address_space(3))) float*)&smem[wv][col];
        asm volatile("global_load_async_to_lds_b128 %0, %1, off"
                     :: "v"(lds_addr), "v"(src + col)
                     : "memory");
#endif
    }

    // Producer and consumer are the same wave: ASYNCcnt wait is the only
    // ordering needed before reading LDS (no s_barrier).
#if __has_builtin(__builtin_amdgcn_s_wait_asynccnt)
    __builtin_amdgcn_s_wait_asynccnt(0);
#else
    asm volatile("s_wait_asynccnt 0" ::: "memory");
#endif

    // ---- pull row back (ds_load_b128) and accumulate sum / sum^2 ----------
    f32x4 v[CHUNKS];
    float sum = 0.0f, ssq = 0.0f;
    #pragma unroll
    for (int c = 0; c < CHUNKS; ++c) {
        v[c] = *(const f32x4*)&smem[wv][c * (WAVE * 4) + lane * 4];
        sum += v[c].x + v[c].y + v[c].z + v[c].w;
        ssq += v[c].x * v[c].x + v[c].y * v[c].y
             + v[c].z * v[c].z + v[c].w * v[c].w;
    }

    // ---- wave32 butterfly reduction (5 steps, NOT 6 -- this is wave32) ----
    #pragma unroll
    for (int m = WAVE / 2; m > 0; m >>= 1) {
        sum += __shfl_xor(sum, m, WAVE);
        ssq += __shfl_xor(ssq, m, WAVE);
    }

    const float inv_w = 1.0f / (float)ROW_W;
    const float mean  = sum * inv_w;
    const float var   = ssq * inv_w - mean * mean;   // == E[(x-mean)^2]
    const float rstd  = rsqrtf(var + ROW_EPS);       // == exp(-0.5*log(var+eps))

    // ---- normalize from registers, coalesced b128 stores ------------------
    float* dst = y + (size_t)row * ROW_W;
    #pragma unroll
    for (int c = 0; c < CHUNKS; ++c) {
        f32x4 o;
        o.x = (v[c].x - mean) * rstd;
        o.y = (v[c].y - mean) * rstd;
        o.z = (v[c].z - mean) * rstd;
        o.w = (v[c].w - mean) * rstd;
        *(f32x4*)(dst + c * (WAVE * 4) + lane * 4) = o;
    }
}

extern "C" void kernel_launch(void* const* d_in, const int* in_sizes, int n_in,
                              void* d_out, int out_size, void* d_ws, size_t ws_size,
                              hipStream_t stream) {
    const float* x = (const float*)d_in[0];
    float*       y = (float*)d_out;

    const int total = in_sizes[0];           // 33554432
    const int nrows = total / ROW_W;         // 32768

    dim3 block(ROWS_PER_BLOCK * WAVE);       // 256 threads = 8 waves
    dim3 grid((nrows + ROWS_PER_BLOCK - 1) / ROWS_PER_BLOCK);
    rownorm_async_kernel<<<grid, block, 0, stream>>>(x, y, nrows);
}